// ClassificationHeadCRF_41360535061220
// MI455X (gfx1250) — compile-verified
//
#include <hip/hip_runtime.h>
#include <hip/hip_bf16.h>

typedef __attribute__((ext_vector_type(2))) float v2f;
typedef __attribute__((ext_vector_type(8))) float v8f;

#define HID 2048
#define NL  48
#define NB  16
#define SL  512
#define M_TOT   (NB * SL)     // 8192 rows
#define KGROUPS (HID / 4)     // 512 WMMA K-steps

#if defined(__HIP_DEVICE_COMPILE__) && !__has_builtin(__builtin_amdgcn_wmma_f32_16x16x4_f32)
#error "missing __builtin_amdgcn_wmma_f32_16x16x4_f32 on this toolchain"
#endif

// ---------------------------------------------------------------------------
// Kernel 0: repack W2 [2048 x 48] into WMMA B-fragment order so the GEMM
// inner loop does one coalesced b64 load per fragment.
// Fragment layout (4x16 B, mirrors documented A layout): lane L holds
// col n=L&15, K-pair kb=(L>>4)*2 -> two consecutive floats per lane.
// ---------------------------------------------------------------------------
__global__ __launch_bounds__(256) void repack_w2(const float* __restrict__ W2,
                                                 float* __restrict__ W2p) {
    int idx = blockIdx.x * 256 + threadIdx.x;      // [0, 512*3*32)
    if (idx >= KGROUPS * 3 * 32) return;
    int lane = idx & 31;
    int nt   = (idx >> 5) % 3;                     // n-tile 0..2
    int kg   = idx / 96;                           // K-group
    int n    = lane & 15;
    int kb   = (lane >> 4) * 2;
    int k    = kg * 4 + kb;
    W2p[2 * idx + 0] = W2[(k + 0) * NL + nt * 16 + n];
    W2p[2 * idx + 1] = W2[(k + 1) * NL + nt * 16 + n];
}

// ---------------------------------------------------------------------------
// Kernel 1: logits = relu(H) @ W2 + b2 via V_WMMA_F32_16X16X4_F32.
// One wave per 16-row tile; 3 accumulators cover all 48 output columns.
// ---------------------------------------------------------------------------
__global__ __launch_bounds__(32) void logits_gemm(const float* __restrict__ H,
                                                  const float* __restrict__ W2p,
                                                  const float* __restrict__ b2,
                                                  float* __restrict__ logits) {
    const int mtile = blockIdx.x;                  // 0..511
    const int lane  = threadIdx.x;                 // 0..31
    const int m     = lane & 15;
    const int kb    = (lane >> 4) * 2;             // 0 or 2
    const float* a_ptr = H + (size_t)(mtile * 16 + m) * HID + kb;
    const v2f*   Bp    = (const v2f*)W2p;

    v8f acc0 = {}; v8f acc1 = {}; v8f acc2 = {};
    for (int kg = 0; kg < KGROUPS; ++kg) {
        v2f a = *(const v2f*)(a_ptr + kg * 4);     // 8B contiguous per lane
        a.x = fmaxf(a.x, 0.0f);                    // fused ReLU
        a.y = fmaxf(a.y, 0.0f);
        v2f b0 = Bp[(kg * 3 + 0) * 32 + lane];
        v2f b1 = Bp[(kg * 3 + 1) * 32 + lane];
        v2f b2v= Bp[(kg * 3 + 2) * 32 + lane];
        acc0 = __builtin_amdgcn_wmma_f32_16x16x4_f32(false, a, false, b0,
                                                     (short)0, acc0, false, false);
        acc1 = __builtin_amdgcn_wmma_f32_16x16x4_f32(false, a, false, b1,
                                                     (short)0, acc1, false, false);
        acc2 = __builtin_amdgcn_wmma_f32_16x16x4_f32(false, a, false, b2v,
                                                     (short)0, acc2, false, false);
    }

    // C/D layout: VGPR r -> row (r + 8*(lane>>4)), col (lane&15)
    const int hi = lane >> 4;
    const int n  = lane & 15;
    const float bias0 = b2[n], bias1 = b2[16 + n], bias2 = b2[32 + n];
    float* out = logits + (size_t)mtile * 16 * NL;
#pragma unroll
    for (int r = 0; r < 8; ++r) {
        int mr = r + hi * 8;
        out[mr * NL + n]      = acc0[r] + bias0;
        out[mr * NL + 16 + n] = acc1[r] + bias1;
        out[mr * NL + 32 + n] = acc2[r] + bias2;
    }
}

// ---------------------------------------------------------------------------
// Kernel 2: per-batch CRF forward recurrence + gold path score.
// 192 threads: 4 threads per destination label j; each reduces 12 source
// labels (transition column held in registers), combined with 2 shfl_xor.
// Double-buffered alpha in LDS, one barrier per time step.
// ---------------------------------------------------------------------------
__global__ __launch_bounds__(192) void crf_forward(const float* __restrict__ logits,
                                                   const int*   __restrict__ labels,
                                                   const float* __restrict__ trans,
                                                   const float* __restrict__ start,
                                                   const float* __restrict__ endv,
                                                   float* __restrict__ partial) {
    __shared__ float Tt[NL * 49];        // Tt[j*49 + i] = trans[i*48 + j] (pad kills conflicts)
    __shared__ float bufA[NL], bufB[NL];
    __shared__ float red[192];

    const int b   = blockIdx.x;
    const int tid = threadIdx.x;
    const float* lg  = logits + (size_t)b * SL * NL;
    const int*   lab = labels + b * SL;

    // Stage transposed transitions.
    for (int x = tid; x < NL * NL; x += 192) {
        int i = x / NL, j = x % NL;
        Tt[j * 49 + i] = trans[x];
    }

    // Gold path score (parallel gather + reduce).
    float g = 0.0f;
    for (int t = tid; t < SL; t += 192)     g += lg[t * NL + lab[t]];
    for (int t = tid; t < SL - 1; t += 192) g += trans[lab[t] * NL + lab[t + 1]];
    if (tid == 0) g += start[lab[0]] + endv[lab[SL - 1]];
    red[tid] = g;

    // alpha_0 = start + logits[:,0]
    if (tid < NL) bufA[tid] = start[tid] + lg[tid];
    __syncthreads();
    if (tid == 0) { float s = 0.0f; for (int i = 0; i < 192; ++i) s += red[i]; red[0] = s; }

    const int j   = tid >> 2;              // 0..47
    const int sub = tid & 3;               // 0..3  (12 source labels each)

    float tr[12];
#pragma unroll
    for (int ii = 0; ii < 12; ++ii) tr[ii] = Tt[j * 49 + sub * 12 + ii];

    float* A  = bufA;
    float* Bn = bufB;
    for (int t = 1; t < SL; ++t) {
        const float* arow = &A[sub * 12];
        float v[12];
        float m = -3.0e38f;
#pragma unroll
        for (int ii = 0; ii < 12; ++ii) { v[ii] = arow[ii] + tr[ii]; m = fmaxf(m, v[ii]); }
        m = fmaxf(m, __shfl_xor(m, 1));
        m = fmaxf(m, __shfl_xor(m, 2));
        float s = 0.0f;
#pragma unroll
        for (int ii = 0; ii < 12; ++ii) s += __expf(v[ii] - m);
        s += __shfl_xor(s, 1);
        s += __shfl_xor(s, 2);
        if (sub == 0) Bn[j] = m + __logf(s) + lg[t * NL + j];
        __syncthreads();
        float* tmp = A; A = Bn; Bn = tmp;
    }

    if (tid == 0) {
        float m = -3.0e38f;
        for (int i = 0; i < NL; ++i) m = fmaxf(m, A[i] + endv[i]);
        float s = 0.0f;
        for (int i = 0; i < NL; ++i) s += __expf(A[i] + endv[i] - m);
        partial[b] = (m + __logf(s)) - red[0];     // logZ - gold
    }
}

// ---------------------------------------------------------------------------
// Kernel 3: deterministic final mean over the 16 batch results.
// ---------------------------------------------------------------------------
__global__ void finalize(const float* __restrict__ partial, float* __restrict__ out) {
    if (threadIdx.x == 0) {
        float s = 0.0f;
        for (int i = 0; i < NB; ++i) s += partial[i];
        out[0] = s / (float)NB;
    }
}

extern "C" void kernel_launch(void* const* d_in, const int* in_sizes, int n_in,
                              void* d_out, int out_size, void* d_ws, size_t ws_size,
                              hipStream_t stream) {
    const float* H     = (const float*)d_in[0];   // [16,512,2048]
    const int*   lab   = (const int*)  d_in[1];   // [16,512]
    const float* W2    = (const float*)d_in[2];   // [2048,48]
    const float* b2    = (const float*)d_in[3];   // [48]
    const float* trans = (const float*)d_in[4];   // [48,48]
    const float* startv= (const float*)d_in[5];   // [48]
    const float* endv  = (const float*)d_in[6];   // [48]

    float* logits  = (float*)d_ws;                          // 8192*48 floats
    float* W2p     = logits + (size_t)M_TOT * NL;           // 98304 floats
    float* partial = W2p + (size_t)KGROUPS * 3 * 32 * 2;    // 16 floats

    repack_w2 <<<(KGROUPS * 3 * 32 + 255) / 256, 256, 0, stream>>>(W2, W2p);
    logits_gemm<<<M_TOT / 16, 32, 0, stream>>>(H, W2p, b2, logits);
    crf_forward<<<NB, 192, 0, stream>>>(logits, lab, trans, startv, endv, partial);
    finalize   <<<1, 32, 0, stream>>>(partial, (float*)d_out);
}